// ImgToBEVDynamic_2860448219533
// MI455X (gfx1250) — compile-verified
//
#include <hip/hip_runtime.h>

typedef float v2f __attribute__((ext_vector_type(2)));
typedef float v8f __attribute__((ext_vector_type(8)));

#define BDIM 4
#define CI   128
#define HI   96
#define WI   320
#define HO   400
#define WO   352

// Per reference: sx = 1408/352 = 4 -> x step = VX*sx = 0.2, XMIN = 0
//                sy = 1600/400 = 4 -> y step = 0.2, YMIN = -40
// px = u/DS, py = v/DS recovered via the same normalize/denormalize chain.

__global__ __launch_bounds__(256) void img2bev_wmma_kernel(
    const float* __restrict__ img,   // (B, Ci, Hi, Wi)
    const float* __restrict__ Kmat,  // (B, 3, 3)
    const float* __restrict__ Tmat,  // (B, 4, 4)
    float* __restrict__ out)         // (B, Ci, HO, WO)
{
    __shared__ float lds_cam[32 * 3];   // X,Y,Z for 32 pixels of this tile

    const int tid   = threadIdx.x;
    const int lane  = tid & 31;
    const int wbase = blockIdx.x * 32;
    const int h     = blockIdx.y;
    const int b     = blockIdx.z;

    // ---------------- B operand: B[k][n] = T[n][k] for n<4 (else 0) -------
    // 32-bit 4x16 B layout: VGPR0 = rows K0 (lanes 0-15) / K2 (lanes 16-31),
    //                       VGPR1 = rows K1 / K3.
    const float* Tb = Tmat + b * 16;
    const int n = lane & 15;
    v2f bmat; bmat.x = 0.0f; bmat.y = 0.0f;
    if (n < 4) {
        if (lane < 16) { bmat.x = Tb[n * 4 + 0]; bmat.y = Tb[n * 4 + 1]; }
        else           { bmat.x = Tb[n * 4 + 2]; bmat.y = Tb[n * 4 + 3]; }
    }

    const float ycoord = ((float)h + 0.5f) * 0.2f + (-40.0f);

    // ---------------- two 16-pixel groups: cam = pts x T^T via WMMA -------
    // 32-bit 16x4 A layout: VGPR0 = K0 (lanes 0-15) / K2 (lanes 16-31),
    //                       VGPR1 = K1 / K3.  pts = (x, y, 0, 1).
#pragma unroll
    for (int g = 0; g < 2; ++g) {
        v2f amat;
        if (lane < 16) {
            amat.x = ((float)(wbase + g * 16 + lane) + 0.5f) * 0.2f; // x (K0)
            amat.y = ycoord;                                         // y (K1)
        } else {
            amat.x = 0.0f;                                           // z (K2)
            amat.y = 1.0f;                                           // w (K3)
        }
        v8f cacc = {};
        v8f d = __builtin_amdgcn_wmma_f32_16x16x4_f32(
            false, amat, false, bmat, (short)0, cacc, false, false);

        // D layout: VGPR r holds M=r (lanes 0-15), M=r+8 (lanes 16-31), N=lane&15
        if (n < 3) {                       // only need cam components 0..2
            const int mbase = g * 16 + ((lane >= 16) ? 8 : 0);
#pragma unroll
            for (int r = 0; r < 8; ++r)
                lds_cam[(mbase + r) * 3 + n] = d[r];
        }
    }
    __syncthreads();

    // ---------------- per-pixel projection -> taps + weights --------------
    const float fx = Kmat[b * 9 + 0], cx = Kmat[b * 9 + 2];
    const float fy = Kmat[b * 9 + 4], cy = Kmat[b * 9 + 5];

    const int p = lane;                       // pixel handled by this lane
    const float X = lds_cam[p * 3 + 0];
    const float Y = lds_cam[p * 3 + 1];
    const float Z = fmaxf(lds_cam[p * 3 + 2], 0.001f);

    const float u = fx * (X / Z) + cx;
    const float v = fy * (Y / Z) + cy;
    const float xn = u * 0.25f / (float)(WI - 1) * 2.0f - 1.0f;
    const float yn = v * 0.25f / (float)(HI - 1) * 2.0f - 1.0f;
    const float px = (xn + 1.0f) * 0.5f * (float)(WI - 1);
    const float py = (yn + 1.0f) * 0.5f * (float)(HI - 1);

    const float x0f = floorf(px), y0f = floorf(py);
    const float x1f = x0f + 1.0f, y1f = y0f + 1.0f;
    const float wx1 = px - x0f, wx0 = 1.0f - wx1;
    const float wy1 = py - y0f, wy0 = 1.0f - wy1;

    const bool vx0 = (x0f >= 0.0f) && (x0f <= (float)(WI - 1));
    const bool vx1 = (x1f >= 0.0f) && (x1f <= (float)(WI - 1));
    const bool vy0 = (y0f >= 0.0f) && (y0f <= (float)(HI - 1));
    const bool vy1 = (y1f >= 0.0f) && (y1f <= (float)(HI - 1));

    const int ix0 = (int)fminf(fmaxf(x0f, 0.0f), (float)(WI - 1));
    const int ix1 = (int)fminf(fmaxf(x1f, 0.0f), (float)(WI - 1));
    const int iy0 = (int)fminf(fmaxf(y0f, 0.0f), (float)(HI - 1));
    const int iy1 = (int)fminf(fmaxf(y1f, 0.0f), (float)(HI - 1));

    const float w00 = wx0 * wy0 * (float)(vx0 && vy0);
    const float w01 = wx1 * wy0 * (float)(vx1 && vy0);
    const float w10 = wx0 * wy1 * (float)(vx0 && vy1);
    const float w11 = wx1 * wy1 * (float)(vx1 && vy1);

    const int o00 = iy0 * WI + ix0;
    const int o01 = iy0 * WI + ix1;
    const int o10 = iy1 * WI + ix0;
    const int o11 = iy1 * WI + ix1;

    const bool alive = (w00 + w01 + w10 + w11) > 0.0f;

    // ---------------- channel loop: 8 cgroups x 16 channels ---------------
    const int cg = tid >> 5;                                  // 0..7
    const float* src0 = img +
        ((size_t)b * CI + (size_t)cg * 16) * (size_t)(HI * WI);
    float* dst0 = out +
        (((size_t)b * CI + (size_t)cg * 16) * HO + h) * (size_t)WO + wbase + p;
    const size_t cs_in  = (size_t)(HI * WI);
    const size_t cs_out = (size_t)HO * (size_t)WO;

    if (alive) {
#pragma unroll 4
        for (int k = 0; k < 16; ++k) {
            const float* s = src0 + (size_t)k * cs_in;
            float val = w00 * s[o00] + w01 * s[o01]
                      + w10 * s[o10] + w11 * s[o11];
            dst0[(size_t)k * cs_out] = val;
        }
    } else {
#pragma unroll 4
        for (int k = 0; k < 16; ++k)
            dst0[(size_t)k * cs_out] = 0.0f;
    }
}

extern "C" void kernel_launch(void* const* d_in, const int* in_sizes, int n_in,
                              void* d_out, int out_size, void* d_ws, size_t ws_size,
                              hipStream_t stream) {
    (void)in_sizes; (void)n_in; (void)out_size; (void)d_ws; (void)ws_size;
    const float* img  = (const float*)d_in[0];   // img_feat (4,128,96,320)
    // d_in[1] = target_bev_feat: only defines the (H,W) shape, not read.
    const float* Kmat = (const float*)d_in[2];   // (4,3,3)
    const float* Tmat = (const float*)d_in[3];   // (4,4,4)
    float* out = (float*)d_out;                  // (4,128,400,352)

    dim3 grid(WO / 32, HO, BDIM);                // 11 x 400 x 4 full tiles
    img2bev_wmma_kernel<<<grid, 256, 0, stream>>>(img, Kmat, Tmat, out);
}